// SoftmaxCrossEntropyLoss_37752762531938
// MI455X (gfx1250) — compile-verified
//
#include <hip/hip_runtime.h>
#include <hip/hip_bf16.h>

#ifndef __has_builtin
#define __has_builtin(x) 0
#endif

// Problem constants (from reference)
#define BATCH      8192
#define NCLS       50257
#define TPB        256                   // 8 wave32 per workgroup
#define CHUNK      8192                  // floats per LDS chunk = 32 KB
#define NCHUNK     (NCLS / CHUNK)        // 6 full chunks
#define TAIL_START (NCHUNK * CHUNK)      // 49152
#define OPS_PER_THREAD (CHUNK / TPB)     // 32 async b32 copies / thread / chunk

#if __has_builtin(__builtin_amdgcn_global_load_async_to_lds_b32)
#define HAVE_ASYNC_LDS 1
#else
#define HAVE_ASYNC_LDS 0
#endif

typedef __attribute__((address_space(1))) int* gptr_i32;
typedef __attribute__((address_space(3))) int* lptr_i32;

// ---- async global -> LDS copy of one 32 KB chunk (b32: row base is only 4B aligned) ----
__device__ __forceinline__ void async_copy_chunk(const float* __restrict__ g, float* l) {
    const int t = threadIdx.x;
#if HAVE_ASYNC_LDS
    #pragma unroll
    for (int i = 0; i < OPS_PER_THREAD; ++i) {
        const int idx = t + i * TPB;
        __builtin_amdgcn_global_load_async_to_lds_b32(
            (gptr_i32)(g + idx),
            (lptr_i32)(l + idx),
            /*offset=*/0, /*cpol=*/0);
    }
#else
    #pragma unroll
    for (int i = 0; i < OPS_PER_THREAD; ++i) {
        const int idx = t + i * TPB;
        l[idx] = g[idx];                 // fallback: stage through VGPRs
    }
#endif
}

__device__ __forceinline__ void wait_async_keep_prefetch() {
#if HAVE_ASYNC_LDS
  #if __has_builtin(__builtin_amdgcn_s_wait_asynccnt)
    __builtin_amdgcn_s_wait_asynccnt(OPS_PER_THREAD);   // allow next chunk's 32 in flight
  #else
    asm volatile("s_wait_asynccnt 0x20" ::: "memory");
  #endif
#endif
}

__device__ __forceinline__ void wait_async_all() {
#if HAVE_ASYNC_LDS
  #if __has_builtin(__builtin_amdgcn_s_wait_asynccnt)
    __builtin_amdgcn_s_wait_asynccnt(0);
  #else
    asm volatile("s_wait_asynccnt 0x0" ::: "memory");
  #endif
#endif
}

// One workgroup per row: single-HBM-pass online logsumexp via LDS double buffer.
__global__ void __launch_bounds__(TPB)
ce_row_kernel(const float* __restrict__ x, const int* __restrict__ tgt,
              float* __restrict__ row_loss) {
    __shared__ __align__(16) float buf[2][CHUNK];   // 64 KB double buffer
    __shared__ float red_m[TPB];
    __shared__ float red_s[TPB];

    const int   row = blockIdx.x;
    const int   t   = threadIdx.x;
    const float* __restrict__ xr = x + (size_t)row * (size_t)NCLS;

    float m = -3.402823466e38f;   // running max (per thread)
    float s = 0.0f;               // running sum of exp(x - m)

    async_copy_chunk(xr, buf[0]);

    for (int c = 0; c < NCHUNK; ++c) {
        if (c + 1 < NCHUNK) {
            async_copy_chunk(xr + (size_t)(c + 1) * CHUNK, buf[(c + 1) & 1]);
            wait_async_keep_prefetch();   // in-order completion => current chunk landed
        } else {
            wait_async_all();
        }
        __syncthreads();                  // all waves' copies for chunk c visible

        const float4* __restrict__ b4 = (const float4*)buf[c & 1];

        // pass 1: chunk-local max (float4 LDS reads, conflict-free stride)
        float lm = m;
        #pragma unroll
        for (int i = 0; i < OPS_PER_THREAD / 4; ++i) {   // 8 x float4 = 32 floats
            float4 v = b4[t + i * TPB];
            lm = fmaxf(lm, fmaxf(fmaxf(v.x, v.y), fmaxf(v.z, v.w)));
        }
        // rescale old partial sum to new max, then accumulate exps
        s *= __expf(m - lm);
        m  = lm;
        #pragma unroll
        for (int i = 0; i < OPS_PER_THREAD / 4; ++i) {
            float4 v = b4[t + i * TPB];
            s += __expf(v.x - m) + __expf(v.y - m) + __expf(v.z - m) + __expf(v.w - m);
        }
        __syncthreads();                  // reads done before buffer is overwritten
    }

    // tail: 50257 - 49152 = 1105 elements, direct global loads
    for (int j = TAIL_START + t; j < NCLS; j += TPB) {
        float v  = xr[j];
        float nm = fmaxf(m, v);
        s = s * __expf(m - nm) + __expf(v - nm);
        m = nm;
    }

    // deterministic block merge of (m, s) pairs
    red_m[t] = m;
    red_s[t] = s;
    __syncthreads();
    for (int off = TPB / 2; off > 0; off >>= 1) {
        if (t < off) {
            float m1 = red_m[t], s1 = red_s[t];
            float m2 = red_m[t + off], s2 = red_s[t + off];
            float M  = fmaxf(m1, m2);
            red_s[t] = s1 * __expf(m1 - M) + s2 * __expf(m2 - M);
            red_m[t] = M;
        }
        __syncthreads();
    }

    if (t == 0) {
        float lse = red_m[0] + logf(red_s[0]);
        int   tg  = tgt[row];
        row_loss[row] = lse - xr[tg];     // -loglik for this row
    }
}

// Deterministic final mean over 8192 per-row losses (single block, fixed order).
__global__ void __launch_bounds__(256)
reduce_mean_kernel(const float* __restrict__ row_loss, float* __restrict__ out) {
    __shared__ float red[256];
    const int t = threadIdx.x;
    float acc = 0.0f;
    for (int i = t; i < BATCH; i += 256) acc += row_loss[i];
    red[t] = acc;
    __syncthreads();
    for (int off = 128; off > 0; off >>= 1) {
        if (t < off) red[t] += red[t + off];
        __syncthreads();
    }
    if (t == 0) out[0] = red[0] * (1.0f / (float)BATCH);
}

extern "C" void kernel_launch(void* const* d_in, const int* in_sizes, int n_in,
                              void* d_out, int out_size, void* d_ws, size_t ws_size,
                              hipStream_t stream) {
    const float* x   = (const float*)d_in[0];   // [8192, 50257] f32
    const int*   tgt = (const int*)d_in[1];     // [8192] i32
    float* row_loss  = (float*)d_ws;            // 8192 floats of scratch
    float* out       = (float*)d_out;           // scalar f32

    ce_row_kernel<<<BATCH, TPB, 0, stream>>>(x, tgt, row_loss);
    reduce_mean_kernel<<<1, 256, 0, stream>>>(row_loss, out);
}